// MYNET_16449724745607
// MI455X (gfx1250) — compile-verified
//
#include <hip/hip_runtime.h>
#include <hip/hip_bf16.h>

#define S_LEN 1024
#define B_LEN 64
#define D_LEN 1024
#define H_LEN 512
#define BD (B_LEN * D_LEN)     // 65536
#define LSPAN 32

typedef __attribute__((ext_vector_type(2))) float v2f;
typedef __attribute__((ext_vector_type(8))) float v8f;

__device__ __forceinline__ v8f wmma4(v2f a, v2f b, v8f c) {
    // V_WMMA_F32_16X16X4_F32 : D = A(16x4 f32) * B(4x16 f32) + C(16x16 f32)
    return __builtin_amdgcn_wmma_f32_16x16x4_f32(false, a, false, b, (short)0, c, false, false);
}

__device__ __forceinline__ float gelu_exact(float x) {
    return 0.5f * x * (1.0f + erff(x * 0.70710678118654752f));
}
__device__ __forceinline__ float sigmoidf(float x) {
    return 1.0f / (1.0f + expf(-x));
}

// ---------------- context = mean over S of features : [B,D] ----------------
__global__ __launch_bounds__(256) void ctx_kernel(const float* __restrict__ feat,
                                                  float* __restrict__ ctx) {
    int i = blockIdx.x * 256 + threadIdx.x;          // 0 .. BD-1
    float s = 0.0f;
    for (int t = 0; t < S_LEN; ++t) s += feat[(size_t)t * BD + i];
    ctx[i] = s * (1.0f / (float)S_LEN);
}

// ---------------- span predictor hidden: h1[b][j] = gelu(ctx@W_sp1 + b) ----
__global__ __launch_bounds__(256) void spanh_kernel(const float* __restrict__ ctx,
                                                    const float* __restrict__ W,
                                                    const float* __restrict__ bias,
                                                    float* __restrict__ h1) {
    int idx = blockIdx.x * 256 + threadIdx.x;        // 0 .. B*H-1
    int b = idx >> 9;
    int j = idx & (H_LEN - 1);
    const float* c = ctx + (size_t)b * D_LEN;
    float sum = bias[j];
    for (int k = 0; k < D_LEN; ++k) sum += c[k] * W[(size_t)k * H_LEN + j];
    h1[idx] = gelu_exact(sum);
}

// ---------------- span_w = softmax(h1 @ W_sp2 + b_sp2) : [B,3] -------------
__global__ __launch_bounds__(64) void spanw_kernel(const float* __restrict__ h1,
                                                   const float* __restrict__ W2,
                                                   const float* __restrict__ b2,
                                                   float* __restrict__ spanw) {
    int b = threadIdx.x;                              // one block of 64
    float z[3];
    for (int c = 0; c < 3; ++c) {
        float s = b2[c];
        for (int k = 0; k < H_LEN; ++k) s += h1[(size_t)b * H_LEN + k] * W2[k * 3 + c];
        z[c] = s;
    }
    float m = fmaxf(z[0], fmaxf(z[1], z[2]));
    float e0 = expf(z[0] - m), e1 = expf(z[1] - m), e2 = expf(z[2] - m);
    float t = e0 + e1 + e2;
    spanw[b * 3 + 0] = e0 / t;
    spanw[b * 3 + 1] = e1 / t;
    spanw[b * 3 + 2] = e2 / t;
}

// ------- relevance hidden GEMM: h = gelu([hist|ctx] @ W_r1 + b_r1) ---------
// rows = pass*2048 + l*64 + b  (4096 rows), K = 2048, N = 512.
// One wave computes a 32(M) x 64(N) strip: 2 M-tiles x 4 N-tiles = 8 f32-WMMA
// accumulators; the 4 B fragments per k-step are reused across both M-tiles.
__global__ __launch_bounds__(32) void relh_gemm(const float* __restrict__ feat,
                                                const float* __restrict__ ctx,
                                                const float* __restrict__ Wr1,
                                                const float* __restrict__ br1,
                                                float* __restrict__ hbuf) {
    const int lane  = threadIdx.x;
    const int strip = blockIdx.x;                 // 1024 strips
    const int nStrips = H_LEN / 64;               // 8
    const int m0    = (strip / nStrips) * 32;
    const int nBase = (strip % nStrips) * 64;

    const int koff = (lane >> 4) * 2;
    const int ncol = nBase + (lane & 15);

    // per-lane A row pointers for the two M-tiles
    const float* A0[2];   // k in [0,1024): history features
    const float* A1[2];   // k in [1024,2048): broadcast context
#pragma unroll
    for (int mt = 0; mt < 2; ++mt) {
        const int row  = m0 + mt * 16 + (lane & 15);
        const int pass = row >> 11;
        const int r    = row & 2047;
        const int l    = r >> 6;
        const int b    = r & 63;
        const int s    = (pass == 0) ? (992 + l) : (32 * l);
        A0[mt] = feat + (size_t)s * BD + (size_t)b * D_LEN;
        A1[mt] = ctx + (size_t)b * D_LEN;
    }

    v8f acc[2][4] = {};
    // first K half: history features
    for (int k = 0; k < D_LEN; k += 4) {
        const int k0 = k + koff;
        v2f a0 = { A0[0][k0], A0[0][k0 + 1] };
        v2f a1 = { A0[1][k0], A0[1][k0 + 1] };
        const float* W0 = Wr1 + (size_t)k0 * H_LEN + ncol;
#pragma unroll
        for (int t = 0; t < 4; ++t) {
            v2f bv = { W0[16 * t], W0[H_LEN + 16 * t] };
            acc[0][t] = wmma4(a0, bv, acc[0][t]);
            acc[1][t] = wmma4(a1, bv, acc[1][t]);
        }
    }
    // second K half: broadcast context
    for (int k = 0; k < D_LEN; k += 4) {
        const int k0 = k + koff;
        v2f a0 = { A1[0][k0], A1[0][k0 + 1] };
        v2f a1 = { A1[1][k0], A1[1][k0 + 1] };
        const float* W0 = Wr1 + (size_t)(k0 + D_LEN) * H_LEN + ncol;
#pragma unroll
        for (int t = 0; t < 4; ++t) {
            v2f bv = { W0[16 * t], W0[H_LEN + 16 * t] };
            acc[0][t] = wmma4(a0, bv, acc[0][t]);
            acc[1][t] = wmma4(a1, bv, acc[1][t]);
        }
    }
    // C layout: VGPR v -> row base + v + 8*(lane>=16), col nBase+16t+(lane&15)
#pragma unroll
    for (int mt = 0; mt < 2; ++mt) {
#pragma unroll
        for (int v = 0; v < 8; ++v) {
            const int orow = m0 + mt * 16 + v + (lane >> 4) * 8;
#pragma unroll
            for (int t = 0; t < 4; ++t) {
                const int oc = nBase + 16 * t + (lane & 15);
                hbuf[(size_t)orow * H_LEN + oc] = gelu_exact(acc[mt][t][v] + br1[oc]);
            }
        }
    }
}

// --------- scores[row] = sigmoid(h[row,:] . W_r2 + b_r2), one wave/row -----
__global__ __launch_bounds__(256) void scores_kernel(const float* __restrict__ hbuf,
                                                     const float* __restrict__ Wr2,
                                                     const float* __restrict__ br2,
                                                     float* __restrict__ scores) {
    const int row  = blockIdx.x * 8 + (threadIdx.x >> 5);
    const int lane = threadIdx.x & 31;
    const float* h = hbuf + (size_t)row * H_LEN;
    float sum = 0.0f;
    for (int k = lane; k < H_LEN; k += 32) sum += h[k] * Wr2[k];
#pragma unroll
    for (int off = 16; off > 0; off >>= 1) sum += __shfl_down(sum, off, 32);
    if (lane == 0) scores[row] = sigmoidf(sum + br2[0]);
}

// --------- pooled[pass][b][d] = sum_l softmax_l(score+log(decay))*hist -----
__global__ __launch_bounds__(256) void pool_kernel(const float* __restrict__ feat,
                                                   const float* __restrict__ scores,
                                                   const float* __restrict__ decL,
                                                   const float* __restrict__ decG,
                                                   float* __restrict__ pooled) {
    const int pass = blockIdx.x >> 6;
    const int b    = blockIdx.x & 63;
    __shared__ float w[LSPAN];
    if (threadIdx.x == 0) {
        const float decay = (pass == 0) ? decL[0] : decG[0];
        float e[LSPAN];
        float m = -1e30f;
        for (int l = 0; l < LSPAN; ++l) {
            float d = powf(decay, (float)(LSPAN - 1 - l));
            e[l] = scores[pass * 2048 + l * 64 + b] + logf(d + 1e-8f);
            m = fmaxf(m, e[l]);
        }
        float ssum = 0.0f;
        for (int l = 0; l < LSPAN; ++l) { e[l] = expf(e[l] - m); ssum += e[l]; }
        for (int l = 0; l < LSPAN; ++l) w[l] = e[l] / ssum;
    }
    __syncthreads();
    for (int d = threadIdx.x; d < D_LEN; d += 256) {
        float acc = 0.0f;
        for (int l = 0; l < LSPAN; ++l) {
            const int s = (pass == 0) ? (992 + l) : (32 * l);
            acc += w[l] * feat[(size_t)s * BD + (size_t)b * D_LEN + d];
        }
        pooled[(size_t)pass * BD + (size_t)b * D_LEN + d] = acc;
    }
}

// ---- fusion GEMM: fused_pre = gelu([lf*sw1 | gf*sw2] @ W_f + b_f) ---------
// M=64, N=1024, K=2048
__global__ __launch_bounds__(32) void fused_gemm(const float* __restrict__ pooled,
                                                 const float* __restrict__ spanw,
                                                 const float* __restrict__ Wf,
                                                 const float* __restrict__ bf,
                                                 float* __restrict__ fusedpre) {
    const int lane  = threadIdx.x;
    const int strip = blockIdx.x;               // 4 mtiles * 16 nstrips = 64
    const int m0    = (strip / 16) * 16;
    const int nBase = (strip % 16) * 64;

    const int b   = m0 + (lane & 15);
    const float sw1 = spanw[b * 3 + 1];
    const float sw2 = spanw[b * 3 + 2];
    const float* A0 = pooled + (size_t)b * D_LEN;        // local pooled
    const float* A1 = pooled + BD + (size_t)b * D_LEN;   // global pooled

    const int koff = (lane >> 4) * 2;
    const int ncol = nBase + (lane & 15);

    v8f acc[4] = {};
    for (int k = 0; k < D_LEN; k += 4) {
        const int k0 = k + koff;
        v2f a = { A0[k0] * sw1, A0[k0 + 1] * sw1 };
        const float* W0 = Wf + (size_t)k0 * D_LEN + ncol;
#pragma unroll
        for (int t = 0; t < 4; ++t) {
            v2f bv = { W0[16 * t], W0[D_LEN + 16 * t] };
            acc[t] = wmma4(a, bv, acc[t]);
        }
    }
    for (int k = 0; k < D_LEN; k += 4) {
        const int k0 = k + koff;
        v2f a = { A1[k0] * sw2, A1[k0 + 1] * sw2 };
        const float* W0 = Wf + (size_t)(k0 + D_LEN) * D_LEN + ncol;
#pragma unroll
        for (int t = 0; t < 4; ++t) {
            v2f bv = { W0[16 * t], W0[D_LEN + 16 * t] };
            acc[t] = wmma4(a, bv, acc[t]);
        }
    }
#pragma unroll
    for (int v = 0; v < 8; ++v) {
        const int orow = m0 + v + (lane >> 4) * 8;
#pragma unroll
        for (int t = 0; t < 4; ++t) {
            const int oc = nBase + 16 * t + (lane & 15);
            fusedpre[(size_t)orow * D_LEN + oc] = gelu_exact(acc[t][v] + bf[oc]);
        }
    }
}

// --------- LayerNorm over 64 rows of length D (fused branch) ---------------
__global__ __launch_bounds__(256) void ln_rows_kernel(const float* __restrict__ x,
                                                      const float* __restrict__ g,
                                                      const float* __restrict__ be,
                                                      float* __restrict__ y) {
    const int row = blockIdx.x;
    const float* xr = x + (size_t)row * D_LEN;
    float v[4], sum = 0.0f, ssq = 0.0f;
#pragma unroll
    for (int i = 0; i < 4; ++i) {
        const int d = threadIdx.x + 256 * i;
        v[i] = xr[d];
        sum += v[i];
        ssq += v[i] * v[i];
    }
    __shared__ float rs[256], rq[256];
    rs[threadIdx.x] = sum; rq[threadIdx.x] = ssq;
    __syncthreads();
    for (int off = 128; off > 0; off >>= 1) {
        if (threadIdx.x < off) { rs[threadIdx.x] += rs[threadIdx.x + off];
                                 rq[threadIdx.x] += rq[threadIdx.x + off]; }
        __syncthreads();
    }
    const float mu  = rs[0] * (1.0f / (float)D_LEN);
    const float var = rq[0] * (1.0f / (float)D_LEN) - mu * mu;
    const float inv = rsqrtf(var + 1e-5f);
#pragma unroll
    for (int i = 0; i < 4; ++i) {
        const int d = threadIdx.x + 256 * i;
        y[(size_t)row * D_LEN + d] = (v[i] - mu) * inv * g[d] + be[d];
    }
}

// ---- gate GEMM: gated[b] = fused[b] * sigmoid(fused @ W_g + b_g) ----------
// M=64, N=1024, K=1024 (broadcast over S handled implicitly)
__global__ __launch_bounds__(32) void gate_gemm(const float* __restrict__ fused,
                                                const float* __restrict__ Wg,
                                                const float* __restrict__ bg,
                                                float* __restrict__ gated) {
    const int lane  = threadIdx.x;
    const int strip = blockIdx.x;               // 4 * 16 = 64
    const int m0    = (strip / 16) * 16;
    const int nBase = (strip % 16) * 64;

    const int b = m0 + (lane & 15);
    const float* A0 = fused + (size_t)b * D_LEN;
    const int koff = (lane >> 4) * 2;
    const int ncol = nBase + (lane & 15);

    v8f acc[4] = {};
    for (int k = 0; k < D_LEN; k += 4) {
        const int k0 = k + koff;
        v2f a = { A0[k0], A0[k0 + 1] };
        const float* W0 = Wg + (size_t)k0 * D_LEN + ncol;
#pragma unroll
        for (int t = 0; t < 4; ++t) {
            v2f bv = { W0[16 * t], W0[D_LEN + 16 * t] };
            acc[t] = wmma4(a, bv, acc[t]);
        }
    }
#pragma unroll
    for (int v = 0; v < 8; ++v) {
        const int orow = m0 + v + (lane >> 4) * 8;
#pragma unroll
        for (int t = 0; t < 4; ++t) {
            const int oc = nBase + 16 * t + (lane & 15);
            const float f = fused[(size_t)orow * D_LEN + oc];
            gated[(size_t)orow * D_LEN + oc] = f * sigmoidf(acc[t][v] + bg[oc]);
        }
    }
}

// ---- final: out[s,b,:] = LN(gated[b,:] + features[s,b,:]); attn = 1/S -----
// float4 (b128) loads/stores: this kernel carries ~512 MB of HBM traffic and
// sets the end-to-end floor (~22 us at 23.3 TB/s).
__global__ __launch_bounds__(256) void final_kernel(const float* __restrict__ feat,
                                                    const float* __restrict__ gated,
                                                    const float* __restrict__ gln,
                                                    const float* __restrict__ bln,
                                                    float* __restrict__ out,
                                                    float* __restrict__ attn) {
    const int row = blockIdx.x;                  // s*B + b
    const int b   = row & 63;
    const float4* fr4 = (const float4*)(feat + (size_t)row * D_LEN);
    const float4* gr4 = (const float4*)(gated + (size_t)b * D_LEN);
    const float4 f = fr4[threadIdx.x];
    const float4 g = gr4[threadIdx.x];
    float4 x;
    x.x = f.x + g.x; x.y = f.y + g.y; x.z = f.z + g.z; x.w = f.w + g.w;
    float sum = x.x + x.y + x.z + x.w;
    float ssq = x.x * x.x + x.y * x.y + x.z * x.z + x.w * x.w;

    __shared__ float rs[256], rq[256];
    rs[threadIdx.x] = sum; rq[threadIdx.x] = ssq;
    __syncthreads();
    for (int off = 128; off > 0; off >>= 1) {
        if (threadIdx.x < off) { rs[threadIdx.x] += rs[threadIdx.x + off];
                                 rq[threadIdx.x] += rq[threadIdx.x + off]; }
        __syncthreads();
    }
    const float mu  = rs[0] * (1.0f / (float)D_LEN);
    const float var = rq[0] * (1.0f / (float)D_LEN) - mu * mu;
    const float inv = rsqrtf(var + 1e-5f);

    const float4 gl = ((const float4*)gln)[threadIdx.x];
    const float4 bl = ((const float4*)bln)[threadIdx.x];
    float4 o;
    o.x = (x.x - mu) * inv * gl.x + bl.x;
    o.y = (x.y - mu) * inv * gl.y + bl.y;
    o.z = (x.z - mu) * inv * gl.z + bl.z;
    o.w = (x.w - mu) * inv * gl.w + bl.w;
    ((float4*)(out + (size_t)row * D_LEN))[threadIdx.x] = o;
    if (threadIdx.x == 0) attn[row] = 1.0f / (float)S_LEN;
}

extern "C" void kernel_launch(void* const* d_in, const int* in_sizes, int n_in,
                              void* d_out, int out_size, void* d_ws, size_t ws_size,
                              hipStream_t stream) {
    const float* feat  = (const float*)d_in[0];
    const float* Wsp1  = (const float*)d_in[1];
    const float* bsp1  = (const float*)d_in[2];
    const float* Wsp2  = (const float*)d_in[3];
    const float* bsp2  = (const float*)d_in[4];
    const float* Wr1   = (const float*)d_in[5];
    const float* br1   = (const float*)d_in[6];
    const float* Wr2   = (const float*)d_in[7];
    const float* br2   = (const float*)d_in[8];
    const float* decL  = (const float*)d_in[9];
    const float* decG  = (const float*)d_in[10];
    const float* Wf    = (const float*)d_in[11];
    const float* bf    = (const float*)d_in[12];
    const float* gf    = (const float*)d_in[13];
    const float* bef   = (const float*)d_in[14];
    const float* Wg    = (const float*)d_in[15];
    const float* bg    = (const float*)d_in[16];
    const float* gln   = (const float*)d_in[17];
    const float* bln   = (const float*)d_in[18];

    float* out  = (float*)d_out;
    float* attn = out + (size_t)S_LEN * BD;

    // workspace layout (floats), ~10.1 MB total
    float* ws = (float*)d_ws;
    float* ctx      = ws;                    // 65536
    float* h1       = ctx + BD;              // 32768
    float* spanw    = h1 + B_LEN * H_LEN;    // 192 (rounded to 256)
    float* hbuf     = spanw + 256;           // 4096*512 = 2097152
    float* scores   = hbuf + 4096 * H_LEN;   // 4096
    float* pooled   = scores + 4096;         // 2*65536
    float* fusedpre = pooled + 2 * BD;       // 65536
    float* fused    = fusedpre + BD;         // 65536
    float* gated    = fused + BD;            // 65536

    ctx_kernel<<<BD / 256, 256, 0, stream>>>(feat, ctx);
    spanh_kernel<<<(B_LEN * H_LEN) / 256, 256, 0, stream>>>(ctx, Wsp1, bsp1, h1);
    spanw_kernel<<<1, 64, 0, stream>>>(h1, Wsp2, bsp2, spanw);
    relh_gemm<<<(4096 / 32) * (H_LEN / 64), 32, 0, stream>>>(feat, ctx, Wr1, br1, hbuf);
    scores_kernel<<<4096 / 8, 256, 0, stream>>>(hbuf, Wr2, br2, scores);
    pool_kernel<<<2 * B_LEN, 256, 0, stream>>>(feat, scores, decL, decG, pooled);
    fused_gemm<<<(B_LEN / 16) * (D_LEN / 64), 32, 0, stream>>>(pooled, spanw, Wf, bf, fusedpre);
    ln_rows_kernel<<<B_LEN, 256, 0, stream>>>(fusedpre, gf, bef, fused);
    gate_gemm<<<(B_LEN / 16) * (D_LEN / 64), 32, 0, stream>>>(fused, Wg, bg, gated);
    final_kernel<<<S_LEN * B_LEN, 256, 0, stream>>>(feat, gated, gln, bln, out, attn);
}